// ThreeStateEquilibrium_1709396983973
// MI455X (gfx1250) — compile-verified
//
#include <hip/hip_runtime.h>

// x(10) = R(h*M)^100 * x0 for the linear 3-state ODE, R = Dopri5 stability
// polynomial. Per-lane fp32 VALU; memory-bound (~84 MB @ 23.3 TB/s ~ 3.6 us).

#define DT 0.1f

__device__ __forceinline__ void mm3(const float* __restrict__ A,
                                    const float* __restrict__ B,
                                    float* __restrict__ C) {
#pragma unroll
    for (int i = 0; i < 3; ++i) {
#pragma unroll
        for (int j = 0; j < 3; ++j) {
            float s = A[i * 3 + 0] * B[0 * 3 + j];
            s = __builtin_fmaf(A[i * 3 + 1], B[1 * 3 + j], s);
            s = __builtin_fmaf(A[i * 3 + 2], B[2 * 3 + j], s);
            C[i * 3 + j] = s;
        }
    }
}

__global__ __launch_bounds__(256) void ode_matpow_kernel(
    const float4* __restrict__ args, float* __restrict__ out, int n) {
    int i = blockIdx.x * 256 + threadIdx.x;
    if (i >= n) return;

    // gfx1250 speculative prefetch (global_prefetch_b8) a few waves ahead of
    // the streaming read; clamped so the address stays inside the buffer.
    int pf = i + 64 * 256;
    if (pf >= n) pf = n - 1;
    __builtin_prefetch(&args[pf], 0, 0);

    float4 a = args[i];                 // global_load_b128, coalesced
    float ef = __expf(-a.y);            // e_f = exp(-dg_df)
    float eb = __expf(a.z);             // e_b = exp(+dg_db)

    // Z = h*M
    float Z[9] = { -DT * ef, DT,        0.0f,
                    DT * ef, -2.0f * DT, DT * eb,
                    0.0f,    DT,        -DT * eb };

    // S = I + Z + Z^2/2 + Z^3/6 + Z^4/24 + Z^5/120 + Z^6/600  (Horner)
    const float c6 = 1.0f / 600.0f, c5 = 1.0f / 120.0f, c4 = 1.0f / 24.0f;
    const float c3 = 1.0f / 6.0f,   c2 = 0.5f;
    float S[9], T[9];
#pragma unroll
    for (int k = 0; k < 9; ++k) T[k] = c6 * Z[k];
    T[0] += c5; T[4] += c5; T[8] += c5;
    mm3(Z, T, S); S[0] += c4; S[4] += c4; S[8] += c4;
    mm3(Z, S, T); T[0] += c3; T[4] += c3; T[8] += c3;
    mm3(Z, T, S); S[0] += c2; S[4] += c2; S[8] += c2;
    mm3(Z, S, T); T[0] += 1.0f; T[4] += 1.0f; T[8] += 1.0f;
    mm3(Z, T, S); S[0] += 1.0f; S[4] += 1.0f; S[8] += 1.0f;

    // R = S^100, binary powering: 100 = 0b1100100
    float R[9], U[9];
#pragma unroll
    for (int k = 0; k < 9; ++k) R[k] = S[k];   // S^1
    mm3(R, R, U); mm3(U, S, R);                // bit 1 -> S^3
    mm3(R, R, U);                              // bit 0 -> S^6
    mm3(U, U, R);                              // bit 0 -> S^12
    mm3(R, R, U); mm3(U, S, R);                // bit 1 -> S^25
    mm3(R, R, U);                              // bit 0 -> S^50
    mm3(U, U, R);                              // bit 0 -> S^100

    // x0 = (1/3,1/3,1/3); output = x_b = row 2 of R dotted with x0
    out[i] = (R[6] + R[7] + R[8]) * (1.0f / 3.0f);
}

extern "C" void kernel_launch(void* const* d_in, const int* in_sizes, int n_in,
                              void* d_out, int out_size, void* d_ws, size_t ws_size,
                              hipStream_t stream) {
    const float4* args = (const float4*)d_in[0];  // [B,4] float32
    float* out = (float*)d_out;                   // [B,1] float32
    int n = in_sizes[0] / 4;                      // B = 4,194,304
    int blocks = (n + 255) / 256;
    ode_matpow_kernel<<<blocks, 256, 0, stream>>>(args, out, n);
}